// FieldBuilder_6502580486147
// MI455X (gfx1250) — compile-verified
//
#include <hip/hip_runtime.h>

typedef __attribute__((ext_vector_type(2))) float v2f;
typedef __attribute__((ext_vector_type(8))) float v8f;

#define NMESH 160
#define NMESH2 (NMESH * NMESH)          // 25600
#define NMESH3 (NMESH * NMESH * NMESH)  // 4096000

// Reference weight for corner p (bits: p=4a+2b+c -> offsets a,b,c):
// pc = pos*10, idx = rint(pc), l = pc-idx; offset 0 -> weight l, offset 1 -> 1-l.
// `atom` must already be clamped in-bounds; caller masks the result.
__device__ __forceinline__ float cic_weight(const float* __restrict__ pos,
                                            int atom, int p) {
  float qx = pos[3 * atom + 0] * 10.0f;
  float qy = pos[3 * atom + 1] * 10.0f;
  float qz = pos[3 * atom + 2] * 10.0f;
  float lx = qx - rintf(qx);
  float ly = qy - rintf(qy);
  float lz = qz - rintf(qz);
  float wx = (p & 4) ? (1.0f - lx) : lx;
  float wy = (p & 2) ? (1.0f - ly) : ly;
  float wz = (p & 1) ? (1.0f - lz) : lz;
  return wx * wy * wz;
}

// One wave handles 4 atoms. A 16x16x4 f32 WMMA computes the four 8x8 outer
// products frac[8] (x) emb[8] as the four quadrants of D:
//   A[8i+p, 2i+j] = frac_{atom(i,j)}[p],  B[2i+j, 8j+q] = emb_{atom(i,j)}[q]
// => D[8i+p, 8j+q] = frac*emb for atom(i,j) = base + 2i + j (k-sets intersect
// in exactly {2i+j}: no cross terms). D layout: lane l, elem v -> row
// v+8*(l/16), col l%16, so lane l owns all 8 corner values of
// (atom = base + l/8, channel = l%8).
__global__ void __launch_bounds__(256)
cic_scatter_wmma(const float* __restrict__ pos,
                 const float* __restrict__ emb,
                 float* __restrict__ out, int n_atoms) {
  const int lane = threadIdx.x & 31;
  const int wave = (int)((blockIdx.x * blockDim.x + threadIdx.x) >> 5);
  const int base = wave * 4;
  const int nc = n_atoms - 1;  // clamp bound (n_atoms >= 1)

  // ---- A (16x4 f32). ISA layout: lanes 0-15 hold row M=lane {K=0 in vgpr0,
  // K=1 in vgpr1}; lanes 16-31 hold row M=lane-16 {K=2, K=3}.
  // Active lanes: 0-7 (rows 0-7, atoms base+0/1) and 24-31 (rows 8-15,
  // atoms base+2/3). Branchless: clamped loads + selects keep EXEC simple.
  v2f A;
  {
    const bool act = (lane < 8) | (lane >= 24);
    const int p = lane & 7;                          // corner index (junk for mid lanes)
    const int pa = base + ((lane >= 24) ? 2 : 0);    // pair base atom
    const float w0 = cic_weight(pos, min(pa, nc), p);
    const float w1 = cic_weight(pos, min(pa + 1, nc), p);
    A.x = (act && pa < n_atoms) ? w0 : 0.0f;
    A.y = (act && pa + 1 < n_atoms) ? w1 : 0.0f;
  }

  // ---- B (4x16 f32). Row-striped: vgpr0 = rows {0 | 2}, vgpr1 = rows {1 | 3}
  // (lanes 0-15 | 16-31), column = lane%16. Row 2i+j is nonzero on cols 8j..8j+7.
  v2f B;
  {
    const int c = lane & 7;
    const int half = lane >> 4;        // 0: rows 0/1, 1: rows 2/3
    const int quad = (lane >> 3) & 1;  // which 8-column block inside the half
    const int a0 = base + 2 * half;    // row 2*half   -> atom(half,0), cols 0-7
    const int a1 = a0 + 1;             // row 2*half+1 -> atom(half,1), cols 8-15
    const float e0 = emb[8 * min(a0, nc) + c];
    const float e1 = emb[8 * min(a1, nc) + c];
    B.x = (quad == 0 && a0 < n_atoms) ? e0 : 0.0f;
    B.y = (quad == 1 && a1 < n_atoms) ? e1 : 0.0f;
  }

  // EXEC is all-ones here (no early returns; tail masking done via zero operands).
  v8f C = {};
  C = __builtin_amdgcn_wmma_f32_16x16x4_f32(false, A, false, B, (short)0, C,
                                            false, false);

  // ---- Scatter: this lane owns corner values v=0..7 of (atom, chan).
  const int atom = base + (lane >> 3);
  const int chan = lane & 7;
  if (atom < n_atoms) {
    float qx = pos[3 * atom + 0] * 10.0f;
    float qy = pos[3 * atom + 1] * 10.0f;
    float qz = pos[3 * atom + 2] * 10.0f;
    int ix = (int)rintf(qx);  // in [0,160]
    int iy = (int)rintf(qy);
    int iz = (int)rintf(qz);
    // Wrapped coordinates for offsets 0 and 1 (inputs in [0,161]).
    int x0 = ix;     if (x0 >= NMESH) x0 -= NMESH;
    int x1 = ix + 1; if (x1 >= NMESH) x1 -= NMESH;
    int y0 = iy;     if (y0 >= NMESH) y0 -= NMESH;
    int y1 = iy + 1; if (y1 >= NMESH) y1 -= NMESH;
    int z0 = iz;     if (z0 >= NMESH) z0 -= NMESH;
    int z1 = iz + 1; if (z1 >= NMESH) z1 -= NMESH;
    // 32-bit partial offsets (max total 8*160^3 = 32.77M elements: fits u32,
    // and the byte offset fits 31 bits -> saddr + voffset addressing).
    const unsigned xp0 = (unsigned)chan * NMESH3 + (unsigned)x0 * NMESH2;
    const unsigned xp1 = (unsigned)chan * NMESH3 + (unsigned)x1 * NMESH2;
    const unsigned yp0 = (unsigned)y0 * NMESH;
    const unsigned yp1 = (unsigned)y1 * NMESH;
#pragma unroll
    for (int v = 0; v < 8; ++v) {
      const unsigned xp = (v & 4) ? xp1 : xp0;
      const unsigned yp = (v & 2) ? yp1 : yp0;
      const unsigned zp = (v & 1) ? (unsigned)z1 : (unsigned)z0;
      atomicAdd(out + (xp + yp + zp), C[v]);  // global_atomic_add_f32, no return
    }
  }
}

extern "C" void kernel_launch(void* const* d_in, const int* in_sizes, int n_in,
                              void* d_out, int out_size, void* d_ws, size_t ws_size,
                              hipStream_t stream) {
  const float* pos = (const float*)d_in[0];  // [N,3] f32
  const float* emb = (const float*)d_in[1];  // [N,8] f32
  float* out = (float*)d_out;                // [8,160,160,160] f32
  const int n_atoms = in_sizes[0] / 3;

  // Zero the accumulation target every call (graph-capturable memset node).
  hipMemsetAsync(out, 0, (size_t)out_size * sizeof(float), stream);

  // 4 atoms per wave, 8 waves (256 threads) per block.
  const int waves = (n_atoms + 3) / 4;
  const long threads = (long)waves * 32;
  const int block = 256;
  const int grid = (int)((threads + block - 1) / block);
  cic_scatter_wmma<<<grid, block, 0, stream>>>(pos, emb, out, n_atoms);
}